// LightGCNModel_27882927685659
// MI455X (gfx1250) — compile-verified
//
#include <hip/hip_runtime.h>
#include <hip/hip_bf16.h>
#include <math.h>

// ---- problem constants (match reference) ----
#define USER_NUM 100000
#define ITEM_NUM 50000
#define N_NODES  (USER_NUM + ITEM_NUM)   // 150000
#define EMB      64
#define E_TOT    4000000                 // 2 * E_HALF directed edges per edge-set
#define NM       (N_NODES * EMB)         // 9,600,000 floats
#define N_PAD    150016                  // padded node count (256B-friendly)

typedef float v2f __attribute__((ext_vector_type(2)));
typedef float v8f __attribute__((ext_vector_type(8)));

static constexpr int TPB = 256;

// ---------------- utility kernels ----------------

__global__ void k_zero_u32(unsigned int* __restrict__ p, int n) {
  int i = blockIdx.x * blockDim.x + threadIdx.x;
  if (i < n) p[i] = 0u;
}

__global__ void k_zero_f4(float4* __restrict__ p, int n4) {
  int i = blockIdx.x * blockDim.x + threadIdx.x;
  if (i < n4) p[i] = make_float4(0.f, 0.f, 0.f, 0.f);
}

// degree count (pos edges only; bidirectional graph => deg_out == deg_in)
__global__ void k_count(const int* __restrict__ src, unsigned int* __restrict__ deg, int n) {
  int i = blockIdx.x * blockDim.x + threadIdx.x;
  if (i < n) atomicAdd(&deg[src[i]], 1u);
}

__global__ void k_rsq(const unsigned int* __restrict__ deg, float* __restrict__ rsq, int n) {
  int i = blockIdx.x * blockDim.x + threadIdx.x;
  if (i < n) {
    float d = (float)deg[i];
    if (d < 1.0f) d = 1.0f;
    rsq[i] = 1.0f / sqrtf(d);
  }
}

// x = res = concat(U, I)
__global__ void k_init(const float4* __restrict__ U, const float4* __restrict__ I,
                       float4* __restrict__ x, float4* __restrict__ res,
                       int n4u, int n4) {
  int i = blockIdx.x * blockDim.x + threadIdx.x;
  if (i < n4) {
    float4 v = (i < n4u) ? U[i] : I[i - n4u];
    x[i]   = v;
    res[i] = v;
  }
}

// agg[dst] += x[src] * rsq[src] ; 16 lanes per edge, float4 per lane
__global__ void k_scatter(const float* __restrict__ x, const float* __restrict__ rsq,
                          const int* __restrict__ src, const int* __restrict__ dst,
                          float* __restrict__ agg, int nEdges) {
  int tid  = blockIdx.x * blockDim.x + threadIdx.x;
  int e    = tid >> 4;
  int lane = tid & 15;
  if (e >= nEdges) return;
  int s = src[e];
  int d = dst[e];
  float sc = rsq[s];
  const float4 v = *(const float4*)(x + (size_t)s * EMB + lane * 4);
  float* base = agg + (size_t)d * EMB + lane * 4;
  atomicAdd(base + 0, v.x * sc);
  atomicAdd(base + 1, v.y * sc);
  atomicAdd(base + 2, v.z * sc);
  atomicAdd(base + 3, v.w * sc);
}

// emb = agg * rsq_in ; x <- emb ; res += emb * scale
__global__ void k_combine(const float4* __restrict__ agg, const float* __restrict__ rsq,
                          float4* __restrict__ x, float4* __restrict__ res,
                          float scale, int n4) {
  int i = blockIdx.x * blockDim.x + threadIdx.x;
  if (i >= n4) return;
  int node = i >> 4;            // 16 float4 per 64-float row
  float sc = rsq[node];
  float4 a = agg[i];
  float4 e;
  e.x = a.x * sc; e.y = a.y * sc; e.z = a.z * sc; e.w = a.w * sc;
  x[i] = e;
  float4 r = res[i];
  r.x += e.x * scale; r.y += e.y * scale; r.z += e.z * scale; r.w += e.w * scale;
  res[i] = r;
}

// ---------------- WMMA edge dot products ----------------
// One wave32 handles 16 edges. Accumulate D = A x B over K=64 with 16
// V_WMMA_F32_16X16X4_F32 steps, where A row m = h[src[e0+m]], B col n =
// h[dst[e0+n]]. diag(D) = wanted dot products, in full f32 precision.
//
// A 16x4 f32 layout: lane = M + 16*(K>=2), vgpr = K%2
// B 4x16  f32 layout: lane = N + 16*(K>=2), vgpr = K%2
// => lane l loads float2 at column (kb + 2*(l>>4)) from both its src row and
//    dst row (row index = l&15). Identical bytes/edge to a shuffle-reduce.
__global__ void k_edge_dot_wmma(const float* __restrict__ h,
                                const int* __restrict__ src,
                                const int* __restrict__ dst,
                                float* __restrict__ out, int nEdges) {
  int wave = (blockIdx.x * blockDim.x + threadIdx.x) >> 5;
  int lane = threadIdx.x & 31;
  int e0 = wave * 16;
  if (e0 >= nEdges) return;            // uniform across the wave: EXEC stays all-1s

  int l16 = lane & 15;
  int sel = lane >> 4;                 // 0: K pair {0,1}; 1: K pair {2,3}
  const float* ps = h + (size_t)src[e0 + l16] * EMB;
  const float* pd = h + (size_t)dst[e0 + l16] * EMB;

  v8f c = {};
#pragma unroll
  for (int kb = 0; kb < EMB; kb += 4) {
    int col = kb + 2 * sel;
    v2f a = *(const v2f*)(ps + col);
    v2f b = *(const v2f*)(pd + col);
    // (neg_a, A, neg_b, B, c_mod, C, reuse_a, reuse_b)
    c = __builtin_amdgcn_wmma_f32_16x16x4_f32(false, a, false, b, (short)0, c,
                                              false, false);
  }

  // diag(D): e<8 at (vgpr=e, lane=e); e>=8 at (vgpr=e-8, lane=e+16)
  int idx = -1;
  int e = -1;
  if (lane < 8)        { idx = lane;      e = e0 + lane; }
  else if (lane >= 24) { idx = lane - 24; e = e0 + (lane - 16); }
  if (idx >= 0) {
    float val = c[0];
    if (idx == 1) val = c[1];
    if (idx == 2) val = c[2];
    if (idx == 3) val = c[3];
    if (idx == 4) val = c[4];
    if (idx == 5) val = c[5];
    if (idx == 6) val = c[6];
    if (idx == 7) val = c[7];
    out[e] = val;
  }
}

// ---------------- driver ----------------

extern "C" void kernel_launch(void* const* d_in, const int* in_sizes, int n_in,
                              void* d_out, int out_size, void* d_ws, size_t ws_size,
                              hipStream_t stream) {
  const float* U       = (const float*)d_in[0];   // [100000,64]
  const float* I       = (const float*)d_in[1];   // [50000,64]
  const int*   pos_src = (const int*)d_in[2];     // [4M]
  const int*   pos_dst = (const int*)d_in[3];
  const int*   neg_src = (const int*)d_in[4];
  const int*   neg_dst = (const int*)d_in[5];
  float* out = (float*)d_out;                     // [8M]: pos_pred ++ neg_pred

  // workspace layout (all 256B aligned): ~116.4 MB total
  char* ws = (char*)d_ws;
  size_t off = 0;
  float*        rsq = (float*)(ws + off);        off += (size_t)N_PAD * 4;  // 600,064
  unsigned int* deg = (unsigned int*)(ws + off); off += (size_t)N_PAD * 4;
  float*        x   = (float*)(ws + off);        off += (size_t)NM * 4;     // 38.4 MB
  float*        agg = (float*)(ws + off);        off += (size_t)NM * 4;
  float*        res = (float*)(ws + off);        off += (size_t)NM * 4;
  (void)ws_size; (void)in_sizes; (void)n_in; (void)out_size;

  const int n4  = NM / 4;              // 2,400,000 float4
  const int n4u = USER_NUM * EMB / 4;  // 1,600,000 float4

  // degrees + normalization (pos graph only)
  k_zero_u32<<<(N_PAD + TPB - 1) / TPB, TPB, 0, stream>>>(deg, N_PAD);
  k_count<<<(E_TOT + TPB - 1) / TPB, TPB, 0, stream>>>(pos_src, deg, E_TOT);
  k_rsq<<<(N_NODES + TPB - 1) / TPB, TPB, 0, stream>>>(deg, rsq, N_NODES);

  // x = res = concat(U, I)
  k_init<<<(n4 + TPB - 1) / TPB, TPB, 0, stream>>>((const float4*)U, (const float4*)I,
                                                   (float4*)x, (float4*)res, n4u, n4);

  // 3 LightGCN propagation layers
  for (int layer = 0; layer < 3; ++layer) {
    float scale = 1.0f / (float)(layer + 2);
    k_zero_f4<<<(n4 + TPB - 1) / TPB, TPB, 0, stream>>>((float4*)agg, n4);
    int scatterThreads = E_TOT * 16;   // 64M, 16 lanes/edge
    k_scatter<<<(scatterThreads + TPB - 1) / TPB, TPB, 0, stream>>>(
        x, rsq, pos_src, pos_dst, agg, E_TOT);
    k_combine<<<(n4 + TPB - 1) / TPB, TPB, 0, stream>>>(
        (const float4*)agg, rsq, (float4*)x, (float4*)res, scale, n4);
  }

  // edge dot products on final embeddings h = res (WMMA f32 16x16x4)
  int dotThreads = (E_TOT / 16) * 32;  // 8M threads, exact multiple of TPB
  k_edge_dot_wmma<<<dotThreads / TPB, TPB, 0, stream>>>(res, pos_src, pos_dst,
                                                        out, E_TOT);
  k_edge_dot_wmma<<<dotThreads / TPB, TPB, 0, stream>>>(res, neg_src, neg_dst,
                                                        out + E_TOT, E_TOT);
}